// GAT_4999341933107
// MI455X (gfx1250) — compile-verified
//
#include <hip/hip_runtime.h>
#include <hip/hip_bf16.h>

// ---------------------------------------------------------------------------
// 2-layer GAT (heads=1) + linear/softmax classifier for MI455X (gfx1250).
// GEMMs run on V_WMMA_F32_16X16X4_F32 (full fp32 matrix path: workload is
// bandwidth-bound, ~10 GFLOP total, so fp32 WMMA is free and exact).
// Edge softmax-aggregation uses L2-resident f32 atomics (out buffer 51MB << 192MB L2).
// ---------------------------------------------------------------------------

#define GAT_N    100000
#define GAT_E    1600000
#define GAT_EE   (GAT_E + GAT_N)   // edges + self loops
#define GAT_NIN  256
#define GAT_HID  128
#define GAT_NCLS 3

typedef __attribute__((ext_vector_type(2))) float v2f;
typedef __attribute__((ext_vector_type(8))) float v8f;

// ---------------------------------------------------------------------------
// WMMA fp32 GEMM: C[M x N] = A[M x K] * B[K x N], row-major, one wave per
// 16x16 C tile, K stepped by 4 with V_WMMA_F32_16X16X4_F32.
//
// ISA 7.12.2 layouts (wave32):
//   A 16x4 : lanes 0-15 hold M=lane,  K={k0,k0+1} in v0,v1
//            lanes 16-31 hold M=lane-16, K={k0+2,k0+3} in v0,v1
//   B 4x16 : lanes 0-15 hold N=lane,  K={k0,k0+1} in v0,v1
//            lanes 16-31 hold N=lane-16, K={k0+2,k0+3} in v0,v1
//   C/D    : VGPR i: lanes 0-15 -> M=i, lanes 16-31 -> M=i+8, N=lane&15
// ---------------------------------------------------------------------------
__global__ void __launch_bounds__(128)
gemm16_wmma(const float* __restrict__ A, const float* __restrict__ B,
            float* __restrict__ C, int M, int K, int N) {
  const int wave = blockIdx.x * (blockDim.x >> 5) + (threadIdx.x >> 5);
  const int lane = threadIdx.x & 31;
  const int ntiles = N >> 4;
  const int mt = wave / ntiles;
  const int nt = wave - mt * ntiles;
  if (mt * 16 >= M) return;                 // wave-uniform guard (EXEC stays full)

  const int half = lane >> 4;               // 0: K lo pair, 1: K hi pair
  const int l    = lane & 15;

  const float* __restrict__ arow = A + (size_t)(mt * 16 + l) * K + half * 2;
  const float* __restrict__ bcol = B + (size_t)(half * 2) * N + nt * 16 + l;

  v8f acc = {};
  for (int k0 = 0; k0 < K; k0 += 4) {
    v2f a, b;
    a.x = arow[k0];                         // K = k0 + half*2
    a.y = arow[k0 + 1];                     // K = k0 + half*2 + 1
    b.x = bcol[(size_t)k0 * N];             // same K, column nt*16+l
    b.y = bcol[(size_t)(k0 + 1) * N];
    acc = __builtin_amdgcn_wmma_f32_16x16x4_f32(
        /*neg_a=*/false, a, /*neg_b=*/false, b,
        /*c_mod=*/(short)0, acc, /*reuse_a=*/false, /*reuse_b=*/false);
  }

  const int col = nt * 16 + l;
  float* __restrict__ crow = C + (size_t)(mt * 16 + half * 8) * N + col;
#pragma unroll
  for (int i = 0; i < 8; ++i) crow[(size_t)i * N] = acc[i];
}

// --------------------------- helpers ---------------------------------------

__global__ void __launch_bounds__(256)
fill_zero_f(float* __restrict__ p, int n) {
  int i = blockIdx.x * blockDim.x + threadIdx.x;
  if (i < n) p[i] = 0.0f;
}

__global__ void __launch_bounds__(256)
fill_zero_u(unsigned* __restrict__ p, int n) {
  int i = blockIdx.x * blockDim.x + threadIdx.x;
  if (i < n) p[i] = 0u;
}

// order-preserving float<->uint map so segment-max can use integer atomicMax
__device__ __forceinline__ unsigned f2ord(float f) {
  unsigned u = __float_as_uint(f);
  return (u & 0x80000000u) ? ~u : (u | 0x80000000u);
}
__device__ __forceinline__ float ord2f(unsigned u) {
  return __uint_as_float((u & 0x80000000u) ? (u & 0x7FFFFFFFu) : ~u);
}

// per-node attention scalars: s_src[n] = h[n].a_src, s_dst[n] = h[n].a_dst
__global__ void __launch_bounds__(256)
node_scores(const float* __restrict__ h, const float* __restrict__ a_src,
            const float* __restrict__ a_dst, float* __restrict__ s_src,
            float* __restrict__ s_dst, int n) {
  const int wid  = blockIdx.x * (blockDim.x >> 5) + (threadIdx.x >> 5);
  const int lane = threadIdx.x & 31;
  if (wid >= n) return;
  const float* __restrict__ row = h + (size_t)wid * GAT_HID;
  float ss = 0.f, sd = 0.f;
#pragma unroll
  for (int j = 0; j < GAT_HID / 32; ++j) {
    const int f = lane + 32 * j;
    const float v = row[f];
    ss += v * a_src[f];
    sd += v * a_dst[f];
  }
#pragma unroll
  for (int off = 16; off > 0; off >>= 1) {
    ss += __shfl_xor(ss, off, 32);
    sd += __shfl_xor(sd, off, 32);
  }
  if (lane == 0) { s_src[wid] = ss; s_dst[wid] = sd; }
}

// pass 1: e = leaky_relu(ssrc[src] + sdst[dst]); segment max over dst
__global__ void __launch_bounds__(256)
edge_logit_max(const int* __restrict__ src, const int* __restrict__ dst,
               const float* __restrict__ s_src, const float* __restrict__ s_dst,
               float* __restrict__ elog, unsigned* __restrict__ mkey) {
  int i = blockIdx.x * blockDim.x + threadIdx.x;
  if (i >= GAT_EE) return;
  const int s = (i < GAT_E) ? src[i] : (i - GAT_E);
  const int d = (i < GAT_E) ? dst[i] : (i - GAT_E);
  float e = s_src[s] + s_dst[d];
  e = (e > 0.f) ? e : 0.2f * e;
  elog[i] = e;
  atomicMax(mkey + d, f2ord(e));
}

// pass 2: ex = exp(e - max[dst]); den[dst] += ex   (elog updated in place)
__global__ void __launch_bounds__(256)
edge_exp_sum(const int* __restrict__ dst, const unsigned* __restrict__ mkey,
             float* __restrict__ elog, float* __restrict__ den) {
  int i = blockIdx.x * blockDim.x + threadIdx.x;
  if (i >= GAT_EE) return;
  const int d = (i < GAT_E) ? dst[i] : (i - GAT_E);
  const float ex = expf(elog[i] - ord2f(mkey[d]));
  elog[i] = ex;
  atomicAdd(den + d, ex);
}

// pass 3: hagg[dst] += (ex/den[dst]) * hlin[src], one wave per edge,
// 4 coalesced f32 atomics per lane (dst buffer is L2-resident, 51MB << 192MB).
__global__ void __launch_bounds__(256)
edge_aggregate(const int* __restrict__ src, const int* __restrict__ dst,
               const float* __restrict__ ex, const float* __restrict__ den,
               const float* __restrict__ hlin, float* __restrict__ hagg) {
  const int e    = blockIdx.x * (blockDim.x >> 5) + (threadIdx.x >> 5);
  const int lane = threadIdx.x & 31;
  if (e >= GAT_EE) return;
  const int s = (e < GAT_E) ? src[e] : (e - GAT_E);
  const int d = (e < GAT_E) ? dst[e] : (e - GAT_E);
  const float alpha = ex[e] / den[d];
  const float* __restrict__ hs = hlin + (size_t)s * GAT_HID;
  float* __restrict__ od = hagg + (size_t)d * GAT_HID;
#pragma unroll
  for (int j = 0; j < GAT_HID / 32; ++j) {
    const int f = lane + 32 * j;
    atomicAdd(od + f, alpha * hs[f]);
  }
}

// h = relu(h + b)  (b broadcast over HID features)
__global__ void __launch_bounds__(256)
bias_relu(float* __restrict__ h, const float* __restrict__ b, int n) {
  int i = blockIdx.x * blockDim.x + threadIdx.x;
  if (i >= n) return;
  const float v = h[i] + b[i & (GAT_HID - 1)];
  h[i] = v > 0.f ? v : 0.f;
}

// out = softmax(h @ Wo + bo), Wo staged in LDS, one thread per node
__global__ void __launch_bounds__(128)
classify(const float* __restrict__ h, const float* __restrict__ Wo,
         const float* __restrict__ bo, float* __restrict__ out, int n) {
  __shared__ float w[GAT_HID * GAT_NCLS];
  for (int i = threadIdx.x; i < GAT_HID * GAT_NCLS; i += blockDim.x) w[i] = Wo[i];
  __syncthreads();
  const int nid = blockIdx.x * blockDim.x + threadIdx.x;
  if (nid >= n) return;
  const float* __restrict__ row = h + (size_t)nid * GAT_HID;
  float a0 = bo[0], a1 = bo[1], a2 = bo[2];
#pragma unroll 4
  for (int k = 0; k < GAT_HID; ++k) {
    const float v = row[k];
    a0 += v * w[k * 3 + 0];
    a1 += v * w[k * 3 + 1];
    a2 += v * w[k * 3 + 2];
  }
  const float m  = fmaxf(a0, fmaxf(a1, a2));
  const float e0 = expf(a0 - m), e1 = expf(a1 - m), e2 = expf(a2 - m);
  const float inv = 1.f / (e0 + e1 + e2);
  out[nid * 3 + 0] = e0 * inv;
  out[nid * 3 + 1] = e1 * inv;
  out[nid * 3 + 2] = e2 * inv;
}

// ---------------------------------------------------------------------------

extern "C" void kernel_launch(void* const* d_in, const int* in_sizes, int n_in,
                              void* d_out, int out_size, void* d_ws, size_t ws_size,
                              hipStream_t stream) {
  (void)in_sizes; (void)n_in; (void)out_size; (void)ws_size;

  const float* x    = (const float*)d_in[0];
  const int*   ei   = (const int*)  d_in[1];   // [2, E]: row0 = src, row1 = dst
  const float* W1   = (const float*)d_in[2];
  const float* a1s  = (const float*)d_in[3];
  const float* a1d  = (const float*)d_in[4];
  const float* b1   = (const float*)d_in[5];
  const float* W2   = (const float*)d_in[6];
  const float* a2s  = (const float*)d_in[7];
  const float* a2d  = (const float*)d_in[8];
  const float* b2   = (const float*)d_in[9];
  const float* Wo   = (const float*)d_in[10];
  const float* bo   = (const float*)d_in[11];
  float* out        = (float*)d_out;

  const int* srcv = ei;
  const int* dstv = ei + GAT_E;

  // workspace carve-up (all fp32 unless noted)
  float*    ws   = (float*)d_ws;
  float*    hlin = ws;                               // N x HID  (linear transform)
  float*    hagg = hlin + (size_t)GAT_N * GAT_HID;   // N x HID  (aggregated)
  float*    ssrc = hagg + (size_t)GAT_N * GAT_HID;   // N
  float*    sdst = ssrc + GAT_N;                     // N
  unsigned* mkey = (unsigned*)(sdst + GAT_N);        // N (ordered-uint max)
  float*    den  = (float*)(mkey + GAT_N);           // N
  float*    elog = den + GAT_N;                      // EE (logits, then exp)

  const int nhid       = GAT_N * GAT_HID;
  const dim3 blk256(256), blk128(128);
  const int g_gemm  = (GAT_N / 16) * (GAT_HID / 16) / 4;  // 4 waves/block
  const int g_nhid  = (nhid + 255) / 256;
  const int g_node  = (GAT_N + 7) / 8;                    // 8 waves/block
  const int g_nsm   = (GAT_N + 255) / 256;
  const int g_edge  = (GAT_EE + 255) / 256;
  const int g_eagg  = (GAT_EE + 7) / 8;                   // wave per edge
  const int g_cls   = (GAT_N + 127) / 128;

  // ---------------- layer 1 ----------------
  gemm16_wmma<<<g_gemm, blk128, 0, stream>>>(x, W1, hlin, GAT_N, GAT_NIN, GAT_HID);
  node_scores<<<g_node, blk256, 0, stream>>>(hlin, a1s, a1d, ssrc, sdst, GAT_N);
  fill_zero_f<<<g_nhid, blk256, 0, stream>>>(hagg, nhid);
  fill_zero_u<<<g_nsm,  blk256, 0, stream>>>(mkey, GAT_N);
  fill_zero_f<<<g_nsm,  blk256, 0, stream>>>(den,  GAT_N);
  edge_logit_max<<<g_edge, blk256, 0, stream>>>(srcv, dstv, ssrc, sdst, elog, mkey);
  edge_exp_sum  <<<g_edge, blk256, 0, stream>>>(dstv, mkey, elog, den);
  edge_aggregate<<<g_eagg, blk256, 0, stream>>>(srcv, dstv, elog, den, hlin, hagg);
  bias_relu<<<g_nhid, blk256, 0, stream>>>(hagg, b1, nhid);

  // ---------------- layer 2 ----------------
  gemm16_wmma<<<g_gemm, blk128, 0, stream>>>(hagg, W2, hlin, GAT_N, GAT_HID, GAT_HID);
  node_scores<<<g_node, blk256, 0, stream>>>(hlin, a2s, a2d, ssrc, sdst, GAT_N);
  fill_zero_f<<<g_nhid, blk256, 0, stream>>>(hagg, nhid);   // after gemm consumed it
  fill_zero_u<<<g_nsm,  blk256, 0, stream>>>(mkey, GAT_N);
  fill_zero_f<<<g_nsm,  blk256, 0, stream>>>(den,  GAT_N);
  edge_logit_max<<<g_edge, blk256, 0, stream>>>(srcv, dstv, ssrc, sdst, elog, mkey);
  edge_exp_sum  <<<g_edge, blk256, 0, stream>>>(dstv, mkey, elog, den);
  edge_aggregate<<<g_eagg, blk256, 0, stream>>>(srcv, dstv, elog, den, hlin, hagg);
  bias_relu<<<g_nhid, blk256, 0, stream>>>(hagg, b2, nhid);

  // ---------------- classifier ----------------
  classify<<<g_cls, blk128, 0, stream>>>(hagg, Wo, bo, out, GAT_N);
}